// BoundaryLoss_25769803776314
// MI455X (gfx1250) — compile-verified
//
#include <hip/hip_runtime.h>
#include <math.h>

#define IMG_H 512
#define IMG_W 512
#define TILE  64
#define TROWS 66      // tile + 2 halo
#define TPITCH 68     // padded LDS row pitch (floats) to stagger banks

typedef __attribute__((ext_vector_type(2))) float v2f;
typedef __attribute__((ext_vector_type(8))) float v8f;

// ---------------------------------------------------------------------------
// Kernel 1: per-(image,64x64-tile) Sobel boundary-loss partial sums.
// Stages 66x66 halo tiles of labels & probs into LDS with gfx1250 async
// global->LDS DMA, computes Sobel magnitudes from registers, reduces the
// workgroup to one float written to d_ws (deterministic, no atomics).
// ---------------------------------------------------------------------------
__global__ __launch_bounds__(256) void sobel_tile_kernel(
    const float* __restrict__ probs,
    const float* __restrict__ labels,
    float* __restrict__ partial)
{
    __shared__ float sL[TROWS * TPITCH];
    __shared__ float sP[TROWS * TPITCH];
    __shared__ float red[8];

    const int wg   = blockIdx.x;          // 0..4095
    const int img  = wg >> 6;             // 0..63  (b*4 + (c-1))
    const int tile = wg & 63;             // 0..63  (8x8 tiles)
    const int trow = (tile >> 3) * TILE;
    const int tcol = (tile & 7) * TILE;
    const int b    = img >> 2;
    const int ch   = 1 + (img & 3);
    const size_t base = (size_t)(b * 5 + ch) * (size_t)(IMG_H * IMG_W);
    const float* gL = labels + base;
    const float* gP = probs  + base;

    const int tid = threadIdx.x;

    // ---- stage halo tiles: async DMA for in-range cells, zero the padding ----
    for (int idx = tid; idx < TROWS * TROWS; idx += 256) {
        const int r  = idx / TROWS;
        const int cc = idx - r * TROWS;
        const int gr = trow - 1 + r;
        const int gc = tcol - 1 + cc;
        const int lo = r * TPITCH + cc;
        if ((unsigned)gr < IMG_H && (unsigned)gc < IMG_W) {
            const unsigned ldsL = (unsigned)(unsigned long long)(&sL[lo]);
            const unsigned ldsP = (unsigned)(unsigned long long)(&sP[lo]);
            const float* aL = gL + gr * IMG_W + gc;
            const float* aP = gP + gr * IMG_W + gc;
            asm volatile("global_load_async_to_lds_b32 %0, %1, off"
                         :: "v"(ldsL), "v"(aL) : "memory");
            asm volatile("global_load_async_to_lds_b32 %0, %1, off"
                         :: "v"(ldsP), "v"(aP) : "memory");
        } else {
            sL[lo] = 0.0f;
            sP[lo] = 0.0f;
        }
    }
    asm volatile("s_wait_asynccnt 0" ::: "memory");
    __syncthreads();

    // ---- each thread: 4x4 output block, 3-row sliding register window ----
    const int tx = tid & 15;
    const int ty = tid >> 4;
    const int c0 = tx * 4;                // output col within tile
    const int r0 = ty * 4;                // output row within tile
    // output (r,c) uses LDS rows r..r+2, cols c..c+2 (halo offset folded in)

    float l0[6], l1[6], l2[6], p0[6], p1[6], p2[6];
#pragma unroll
    for (int k = 0; k < 6; ++k) {
        l0[k] = sL[(r0 + 0) * TPITCH + c0 + k];
        l1[k] = sL[(r0 + 1) * TPITCH + c0 + k];
        p0[k] = sP[(r0 + 0) * TPITCH + c0 + k];
        p1[k] = sP[(r0 + 1) * TPITCH + c0 + k];
    }

    float acc = 0.0f;
#pragma unroll
    for (int i = 0; i < 4; ++i) {
#pragma unroll
        for (int k = 0; k < 6; ++k) {
            l2[k] = sL[(r0 + i + 2) * TPITCH + c0 + k];
            p2[k] = sP[(r0 + i + 2) * TPITCH + c0 + k];
        }
#pragma unroll
        for (int c = 0; c < 4; ++c) {
            // labels gradient magnitude
            float hd0 = l0[c] - l0[c + 2];
            float hd1 = l1[c] - l1[c + 2];
            float hd2 = l2[c] - l2[c + 2];
            float gx  = hd0 + 2.0f * hd1 + hd2;
            float hs0 = l0[c] + 2.0f * l0[c + 1] + l0[c + 2];
            float hs2 = l2[c] + 2.0f * l2[c + 1] + l2[c + 2];
            float gy  = hs0 - hs2;
            float Gl  = sqrtf(gx * gx + gy * gy + 1e-6f);
            // probs gradient magnitude
            float qd0 = p0[c] - p0[c + 2];
            float qd1 = p1[c] - p1[c + 2];
            float qd2 = p2[c] - p2[c + 2];
            float qx  = qd0 + 2.0f * qd1 + qd2;
            float qs0 = p0[c] + 2.0f * p0[c + 1] + p0[c + 2];
            float qs2 = p2[c] + 2.0f * p2[c + 1] + p2[c + 2];
            float qy  = qs0 - qs2;
            float Gp  = sqrtf(qx * qx + qy * qy + 1e-6f);
            float d   = Gl - Gp;
            acc += d * d;
        }
#pragma unroll
        for (int k = 0; k < 6; ++k) {
            l0[k] = l1[k]; l1[k] = l2[k];
            p0[k] = p1[k]; p1[k] = p2[k];
        }
    }

    // ---- workgroup reduction (wave32) ----
#pragma unroll
    for (int off = 16; off > 0; off >>= 1)
        acc += __shfl_xor(acc, off, 32);
    if ((tid & 31) == 0) red[tid >> 5] = acc;
    __syncthreads();
    if (tid == 0) {
        float s = 0.0f;
#pragma unroll
        for (int w = 0; w < 8; ++w) s += red[w];
        partial[wg] = s;
    }
}

// ---------------------------------------------------------------------------
// Kernel 2: sum 4096 partials as a chained matmul-accumulate on the matrix
// pipe: D += ones(16x4) * B(4x16-chunk). Every row of D holds the running
// column sums, so sum(d[0]) over 32 lanes == 2 * total.
// ---------------------------------------------------------------------------
__global__ __launch_bounds__(32) void reduce_wmma_kernel(
    const float* __restrict__ part, float* __restrict__ out,
    int n, float scale)
{
    const int lane = threadIdx.x;                       // 0..31
    v2f a; a.x = 1.0f; a.y = 1.0f;                      // A = ones(16x4)
    v8f d = {};                                         // accumulator C/D
    const int col = (lane < 16) ? lane : (lane + 16);   // B[k][n] slot for VGPR0

    for (int chunk = 0; chunk < n; chunk += 64) {
        v2f bvec;
        bvec.x = part[chunk + col];
        bvec.y = part[chunk + col + 16];
        d = __builtin_amdgcn_wmma_f32_16x16x4_f32(
                /*neg_a=*/false, a, /*neg_b=*/false, bvec,
                /*c_mod=*/(short)0, d, /*reuse_a=*/false, /*reuse_b=*/false);
    }

    float s = d[0];                                     // rows 0 and 8 of D
#pragma unroll
    for (int off = 16; off > 0; off >>= 1)
        s += __shfl_xor(s, off, 32);                    // = 2 * total
    if (lane == 0) out[0] = s * 0.5f * scale;
}

extern "C" void kernel_launch(void* const* d_in, const int* in_sizes, int n_in,
                              void* d_out, int out_size, void* d_ws, size_t ws_size,
                              hipStream_t stream)
{
    const float* probs  = (const float*)d_in[0];
    const float* labels = (const float*)d_in[1];
    float* partial = (float*)d_ws;       // 4096 floats = 16 KB scratch
    float* out     = (float*)d_out;

    // 64 images * 64 tiles = 4096 workgroups; all partials rewritten each call.
    sobel_tile_kernel<<<4096, 256, 0, stream>>>(probs, labels, partial);

    const float scale = 1.0f / (16.0f * 512.0f * 512.0f);  // exact power of 2
    reduce_wmma_kernel<<<1, 32, 0, stream>>>(partial, out, 4096, scale);
}